// LearnableFP6Activation_19267223289892
// MI455X (gfx1250) — compile-verified
//
#include <hip/hip_runtime.h>

#ifndef __has_builtin
#define __has_builtin(x) 0
#endif

typedef __attribute__((ext_vector_type(4))) float v4f;
typedef __attribute__((ext_vector_type(2))) int v2i;

#define BLOCK 256
#define LUT_N 64
#define MAX_GRID 16384

// Nearest-codebook quantization against a sorted 64-entry LUT in LDS.
// Top 3 binary-search levels use register-hoisted values (c31, c15/c47,
// c7/c23/c39/c55) -> pure v_cndmask; bottom 3 levels hit LDS; the final
// low/high pair is read through one pointer so the backend merges it into
// a single ds_load_2addr_b32 (offset0:0 offset1:1).
__device__ __forceinline__ float quant64(float xx, const float* lut,
                                         float c31, float c15, float c47,
                                         float c7, float c23, float c39,
                                         float c55) {
  const bool b0 = (c31 < xx);
  const float l1 = b0 ? c47 : c15;
  const bool b1 = (l1 < xx);
  const float t0 = b1 ? c23 : c7;
  const float t1 = b1 ? c55 : c39;
  const float l2 = b0 ? t1 : t0;
  const bool b2 = (l2 < xx);
  int pos = (b0 ? 32 : 0) | (b1 ? 16 : 0) | (b2 ? 8 : 0);
  if (lut[pos + 3] < xx) pos += 4;   // ds_load_b32 (64 entries == 64 banks)
  if (lut[pos + 1] < xx) pos += 2;
  if (lut[pos + 0] < xx) pos += 1;
  // pos == #(vals < xx) == searchsorted(vals, xx, 'left'), in [0, 64]
  const int idx = pos < 1 ? 1 : (pos > 63 ? 63 : pos);   // -> v_med3_i32
  const float* p = lut + (idx - 1);
  const float lo = p[0];   // merged adjacent pair -> ds_load_2addr_b32
  const float hi = p[1];
  // exact reference predicate: ties go to the lower value
  return (__builtin_fabsf(xx - lo) <= __builtin_fabsf(xx - hi)) ? lo : hi;
}

__global__ __launch_bounds__(BLOCK) void fp6_nearest_kernel(
    const float* __restrict__ x, const float* __restrict__ vals,
    float* __restrict__ out, long long n4, long long n) {
  __shared__ float lut[LUT_N];

  // ---- Fill 64-entry LUT (256 B) into LDS ----
#if __has_builtin(__builtin_amdgcn_global_load_async_to_lds_b64)
  // CDNA5 async data path: one wave32, each lane DMAs 8 bytes -> whole LUT
  // in a single global_load_async_to_lds_b64, tracked by ASYNCcnt.
  if (threadIdx.x < 32) {
    const unsigned lane = threadIdx.x;
    __builtin_amdgcn_global_load_async_to_lds_b64(
        (v2i*)(const_cast<float*>(vals) + 2 * lane),
        (v2i*)(&lut[2 * lane]), /*offset=*/0, /*cpol=*/0);
#if __has_builtin(__builtin_amdgcn_s_wait_asynccnt)
    __builtin_amdgcn_s_wait_asynccnt(0);
#else
    asm volatile("s_wait_asynccnt 0" ::: "memory");
#endif
  }
#else
  if (threadIdx.x < LUT_N) lut[threadIdx.x] = vals[threadIdx.x];
#endif
  __syncthreads();

  // Hoist the top 3 search levels into VGPRs (amortized over the whole
  // grid-stride loop; saves 3 of 6 dependent LDS loads per element).
  const float c31 = lut[31];
  const float c15 = lut[15], c47 = lut[47];
  const float c7 = lut[7], c23 = lut[23], c39 = lut[39], c55 = lut[55];

  const v4f* __restrict__ xv = (const v4f*)x;
  v4f* __restrict__ ov = (v4f*)out;
  long long i = (long long)blockIdx.x * BLOCK + threadIdx.x;
  const long long stride = (long long)gridDim.x * BLOCK;

  for (; i < n4; i += stride) {
    if (i + stride < n4)
      __builtin_prefetch(&xv[i + stride], 0, 0);     // global_prefetch_b8
    v4f v = __builtin_nontemporal_load(xv + i);      // global_load_b128, TH=NT
    v4f r;
#pragma unroll
    for (int c = 0; c < 4; ++c)
      r[c] = quant64(v[c], lut, c31, c15, c47, c7, c23, c39, c55);
    __builtin_nontemporal_store(r, ov + i);          // global_store_b128, TH=NT
  }

  // Tail (n not a multiple of 4) — block 0 mops up scalars.
  const long long base = n4 * 4;
  if (blockIdx.x == 0) {
    for (long long j = base + threadIdx.x; j < n; j += BLOCK)
      out[j] = quant64(x[j], lut, c31, c15, c47, c7, c23, c39, c55);
  }
}

extern "C" void kernel_launch(void* const* d_in, const int* in_sizes, int n_in,
                              void* d_out, int out_size, void* d_ws,
                              size_t ws_size, hipStream_t stream) {
  const float* x    = (const float*)d_in[0];   // (8, 4096, 4096) fp32
  const float* vals = (const float*)d_in[1];   // (64,) fp32, sorted ascending
  float* out = (float*)d_out;                  // fp32, same count as x

  const long long n  = (long long)out_size;
  const long long n4 = n >> 2;

  long long blocks = (n4 + BLOCK - 1) / BLOCK;
  if (blocks < 1) blocks = 1;
  if (blocks > MAX_GRID) blocks = MAX_GRID;    // grid-stride: ~8 float4/thread

  fp6_nearest_kernel<<<dim3((unsigned)blocks), dim3(BLOCK), 0, stream>>>(
      x, vals, out, n4, n);
}